// QuadConv_16458314678303
// MI455X (gfx1250) — compile-verified
//
#include <hip/hip_runtime.h>

#define CIN   32
#define COUT  64
#define KNBR  9
#define NW    8            // waves per block (256 threads, 8 wave32)
#define TSTR  36           // padded LDS tile row stride in floats (16B-aligned rows, conflict-free)

typedef float v2f __attribute__((ext_vector_type(2)));
typedef float v4f __attribute__((ext_vector_type(4)));
typedef float v8f __attribute__((ext_vector_type(8)));

__global__ __launch_bounds__(NW * 32) void quadconv_wmma_f32(
    const float* __restrict__ feat,   // [N, 32]
    const int*   __restrict__ nidx,   // [N, 9]  (-1 => zero row)
    const float* __restrict__ W,      // [64, 288] row-major
    const float* __restrict__ bias,   // [64]
    float*       __restrict__ out,    // [N, 64]
    int n)
{
    __shared__ __align__(16) float atile[NW][16 * TSTR];

    const int lane = threadIdx.x & 31;
    const int wave = threadIdx.x >> 5;
    const int base = (blockIdx.x * NW + wave) * 16;   // first node of this wave's tile
    if (base >= n) return;                            // wave-uniform exit

    float* tile = &atile[wave][0];

    // staging roles: 8 lanes per gathered row, float4 per lane
    const int sub = lane & 7;        // which 4-float chunk of the 32-float row
    const int rq  = lane >> 3;       // row quad within each staging step

    // WMMA fragment roles (A 16x4 f32 / B 4x16 f32 layouts)
    const int m  = lane & 15;                 // A: row M / B: column N (within tile)
    const int kh = (lane >> 4) << 1;          // 0 for lanes 0-15, 2 for lanes 16-31

    // per-staging-slot neighbor-list base offsets (node clamped, wave-uniform exit above)
    int nbase[4];
    #pragma unroll
    for (int it = 0; it < 4; ++it) {
        int node = base + it * 4 + rq;
        if (node >= n) node = n - 1;          // harmless duplicate read
        nbase[it] = node * KNBR;
    }

    // hoist bias loads so their latency overlaps the whole kernel
    const float bv0 = bias[m +  0];
    const float bv1 = bias[m + 16];
    const float bv2 = bias[m + 32];
    const float bv3 = bias[m + 48];

    v8f acc0 = (v8f){0.f,0.f,0.f,0.f,0.f,0.f,0.f,0.f};
    v8f acc1 = acc0, acc2 = acc0, acc3 = acc0;

    // ---- two-deep software pipeline: rows(j) in regs, idx(j+1) in regs ----
    int idxc[4], idxn[4];
    v4f rowc[4];

    #pragma unroll
    for (int it = 0; it < 4; ++it)            // idx for j=0
        idxc[it] = nidx[nbase[it] + 0];
    #pragma unroll
    for (int it = 0; it < 4; ++it) {          // rows for j=0 (branch-free clamp)
        const int safe = idxc[it] < 0 ? 0 : idxc[it];
        rowc[it] = *(const v4f*)(feat + (size_t)safe * CIN + sub * 4);
    }
    #pragma unroll
    for (int it = 0; it < 4; ++it)            // idx for j=1
        idxn[it] = nidx[nbase[it] + 1];

    #pragma unroll
    for (int j = 0; j < KNBR; ++j) {
        // ---- commit rows(j) to this wave's LDS tile (zero-select for missing) ----
        #pragma unroll
        for (int it = 0; it < 4; ++it) {
            const int r = it * 4 + rq;
            v4f val = rowc[it];
            if (idxc[it] < 0) val = (v4f){0.f, 0.f, 0.f, 0.f};
            *(v4f*)(tile + r * TSTR + sub * 4) = val;   // ds_store_b128, 16B aligned
        }

        // ---- prefetch rows(j+1) and idx(j+2) before the WMMA block ----
        if (j + 1 < KNBR) {
            #pragma unroll
            for (int it = 0; it < 4; ++it) idxc[it] = idxn[it];
            #pragma unroll
            for (int it = 0; it < 4; ++it) {
                const int safe = idxc[it] < 0 ? 0 : idxc[it];
                rowc[it] = *(const v4f*)(feat + (size_t)safe * CIN + sub * 4);
            }
            if (j + 2 < KNBR) {
                #pragma unroll
                for (int it = 0; it < 4; ++it)
                    idxn[it] = nidx[nbase[it] + j + 2];
            }
        }

        // ---- 8 K-steps of V_WMMA_F32_16X16X4_F32 over this 32-wide K slice ----
        const float* wj = W + j * CIN + kh;               // row n adds n*288
        #pragma unroll
        for (int kk = 0; kk < 8; ++kk) {
            const int k0 = kk * 4;
            // A fragment: lane l -> {A[m][k0+kh], A[m][k0+kh+1]}  (one ds b64)
            const v2f a = *(const v2f*)(tile + m * TSTR + k0 + kh);
            // B fragments: B[k][n] = W[n][k]; lane l -> {W[n][k], W[n][k+1]} (cached global b64)
            const v2f bf0 = *(const v2f*)(wj + (m +  0) * (KNBR * CIN) + k0);
            const v2f bf1 = *(const v2f*)(wj + (m + 16) * (KNBR * CIN) + k0);
            const v2f bf2 = *(const v2f*)(wj + (m + 32) * (KNBR * CIN) + k0);
            const v2f bf3 = *(const v2f*)(wj + (m + 48) * (KNBR * CIN) + k0);
            acc0 = __builtin_amdgcn_wmma_f32_16x16x4_f32(false, a, false, bf0,
                                                         (short)0, acc0, false, false);
            acc1 = __builtin_amdgcn_wmma_f32_16x16x4_f32(false, a, false, bf1,
                                                         (short)0, acc1, false, false);
            acc2 = __builtin_amdgcn_wmma_f32_16x16x4_f32(false, a, false, bf2,
                                                         (short)0, acc2, false, false);
            acc3 = __builtin_amdgcn_wmma_f32_16x16x4_f32(false, a, false, bf3,
                                                         (short)0, acc3, false, false);
        }
    }

    // ---- add bias and store D tiles: lane l, reg v -> node base+v+8*(l/16), chan m+16t ----
    const int rowoff = (lane >> 4) * 8;
    #pragma unroll
    for (int v = 0; v < 8; ++v) {
        const int node = base + rowoff + v;
        if (node < n) {
            float* orow = out + (size_t)node * COUT + m;
            orow[ 0] = acc0[v] + bv0;
            orow[16] = acc1[v] + bv1;
            orow[32] = acc2[v] + bv2;
            orow[48] = acc3[v] + bv3;
        }
    }
}

extern "C" void kernel_launch(void* const* d_in, const int* in_sizes, int n_in,
                              void* d_out, int out_size, void* d_ws, size_t ws_size,
                              hipStream_t stream) {
    (void)n_in; (void)out_size; (void)d_ws; (void)ws_size;
    const float* feat = (const float*)d_in[0];   // [N,32] f32
    const int*   nidx = (const int*)  d_in[1];   // [N,9]  i32 (JAX x64 disabled)
    const float* W    = (const float*)d_in[2];   // [64,288] f32
    const float* b    = (const float*)d_in[3];   // [64] f32
    float*       out  = (float*)d_out;           // [N,64] f32

    const int n      = in_sizes[0] / CIN;                 // 500000
    const int tiles  = (n + 15) / 16;                     // 16-node tiles (waves)
    const int blocks = (tiles + NW - 1) / NW;

    quadconv_wmma_f32<<<blocks, NW * 32, 0, stream>>>(feat, nidx, W, b, out, n);
}